// BiReBlock_3994319586014
// MI455X (gfx1250) — compile-verified
//
#include <hip/hip_runtime.h>

typedef __attribute__((ext_vector_type(2))) float v2f;
typedef __attribute__((ext_vector_type(8))) float v8f;

#define DIN   128
#define DOUT  64
#define EPSV  1e-4f
#define NS_ITERS 18

__device__ __forceinline__ v8f wmma4(v2f a, v2f b, v8f c) {
  // D = A(16x4,f32) * B(4x16,f32) + C(16x16,f32)
  return __builtin_amdgcn_wmma_f32_16x16x4_f32(
      /*neg_a=*/false, a, /*neg_b=*/false, b,
      /*c_mod=*/(short)0, c, /*reuse_a=*/false, /*reuse_b=*/false);
}

// ---------------------------------------------------------------------------
// Kernel A: QR of W^T (128x64) via CGS2 (classical Gram-Schmidt, twice).
// One block, 128 threads. Writes W_st (64x128, row-major) to d_ws.
// ---------------------------------------------------------------------------
__global__ __launch_bounds__(128) void qr_wst_kernel(
    const float* __restrict__ W, float* __restrict__ Wst) {
  __shared__ float Qc[64 * 129];  // column j of W^T at Qc[j*129 + r], padded
  __shared__ float p[64];
  __shared__ float s_rn;
  const int tid = threadIdx.x;  // row index r = tid (0..127)

  // column j of W^T == row j of W (contiguous)
  for (int j = 0; j < 64; ++j) Qc[j * 129 + tid] = W[j * DIN + tid];
  __syncthreads();

  for (int j = 0; j < 64; ++j) {
    for (int pass = 0; pass < 2; ++pass) {
      if (tid < j) {  // thread i computes p[i] = <q_i, v_j>
        float d = 0.f;
        for (int r = 0; r < DIN; ++r)
          d += Qc[tid * 129 + r] * Qc[j * 129 + r];
        p[tid] = d;
      }
      __syncthreads();
      float qv = Qc[j * 129 + tid];
      for (int i = 0; i < j; ++i) qv -= p[i] * Qc[i * 129 + tid];
      Qc[j * 129 + tid] = qv;
      __syncthreads();
    }
    // normalize column j
    float qv = Qc[j * 129 + tid];
    float ss = qv * qv;
    for (int off = 16; off; off >>= 1) ss += __shfl_down(ss, off, 32);
    if ((tid & 31) == 0) p[tid >> 5] = ss;
    __syncthreads();
    if (tid == 0) {
      float t = p[0] + p[1] + p[2] + p[3];
      s_rn = rsqrtf(fmaxf(t, 1e-30f));
    }
    __syncthreads();
    Qc[j * 129 + tid] = qv * s_rn;
    __syncthreads();
  }
  // W_st row o = column o of Q
  for (int j = 0; j < 64; ++j) Wst[j * DIN + tid] = Qc[j * 129 + tid];
}

// ---------------------------------------------------------------------------
// 64x64 GEMM helper: D = A * B, all LDS, row-major 64x64.
// 16 output tiles; wave w owns tiles (w>>2, w&3) and (2+(w>>2), w&3),
// sharing the B fragment (same column tile).
// ---------------------------------------------------------------------------
__device__ __forceinline__ void mm64_lds(const float* __restrict__ A,
                                         const float* __restrict__ Bm,
                                         float* __restrict__ D,
                                         int w, int m, int half) {
  const int mt0 = (w >> 2), mt1 = 2 + (w >> 2), nt = (w & 3);
  const int n = nt * 16 + m;
  v8f acc0 = {}, acc1 = {};
  for (int k = 0; k < 64; k += 4) {
    const int kk = k + 2 * half;
    v2f a, bv;
    bv[0] = Bm[kk * 64 + n];
    bv[1] = Bm[(kk + 1) * 64 + n];
    a[0] = A[(mt0 * 16 + m) * 64 + kk];
    a[1] = A[(mt0 * 16 + m) * 64 + kk + 1];
    acc0 = wmma4(a, bv, acc0);
    a[0] = A[(mt1 * 16 + m) * 64 + kk];
    a[1] = A[(mt1 * 16 + m) * 64 + kk + 1];
    acc1 = wmma4(a, bv, acc1);
  }
  for (int r = 0; r < 8; ++r) {
    D[(mt0 * 16 + r + 8 * half) * 64 + n] = acc0[r];
    D[(mt1 * 16 + r + 8 * half) * 64 + n] = acc1[r];
  }
}

// ---------------------------------------------------------------------------
// Kernel B: fused per-batch  Y = W_st X W^T ; out = re_eig(Y) via Newton-
// Schulz matrix-sign.  grid = 4096 blocks of 256 threads (8 wave32).
// ---------------------------------------------------------------------------
__global__ __launch_bounds__(256) void birsb_kernel(
    const float* __restrict__ X, const float* __restrict__ Wraw,
    const float* __restrict__ Wst, float* __restrict__ out) {
  __shared__ float pool[16384];  // exactly 64 KB, reused across phases

  const int tid = threadIdx.x;
  const int lane = tid & 31;
  const int w = tid >> 5;       // wave 0..7
  const int m = lane & 15;
  const int half = lane >> 4;
  const int b = blockIdx.x;
  const float* __restrict__ Xb = X + (size_t)b * DIN * DIN;

  float* sLo = pool;          // 64x128  (W_st, then W, then A0|S)
  float* sWX = pool + 8192;   // 64x128  (WX, then T|U)

  // ---- stage W_st ----
  for (int i = tid; i < DOUT * DIN; i += 256) sLo[i] = Wst[i];
  __syncthreads();

  // ---- GEMM1: WX(64x128) = W_st(64x128) * X(128x128), X streamed ----
  {
    v8f acc0 = {}, acc1 = {}, acc2 = {}, acc3 = {};
    const int col = w * 16 + m;  // wave w owns column tile nt = w
    for (int k = 0; k < DIN; k += 4) {
      const int kk = k + 2 * half;
      if (k + 4 < DIN)
        __builtin_prefetch(&Xb[(size_t)(kk + 4) * DIN + col], 0, 3);
      v2f bf, a;
      bf[0] = Xb[(size_t)kk * DIN + col];
      bf[1] = Xb[(size_t)(kk + 1) * DIN + col];
      a[0] = sLo[(0 + m) * DIN + kk];  a[1] = sLo[(0 + m) * DIN + kk + 1];
      acc0 = wmma4(a, bf, acc0);
      a[0] = sLo[(16 + m) * DIN + kk]; a[1] = sLo[(16 + m) * DIN + kk + 1];
      acc1 = wmma4(a, bf, acc1);
      a[0] = sLo[(32 + m) * DIN + kk]; a[1] = sLo[(32 + m) * DIN + kk + 1];
      acc2 = wmma4(a, bf, acc2);
      a[0] = sLo[(48 + m) * DIN + kk]; a[1] = sLo[(48 + m) * DIN + kk + 1];
      acc3 = wmma4(a, bf, acc3);
    }
    for (int r = 0; r < 8; ++r) {
      const int rr = r + 8 * half;
      sWX[(0 + rr) * DIN + col] = acc0[r];
      sWX[(16 + rr) * DIN + col] = acc1[r];
      sWX[(32 + rr) * DIN + col] = acc2[r];
      sWX[(48 + rr) * DIN + col] = acc3[r];
    }
  }
  __syncthreads();

  // ---- stage raw W (overwrites W_st slot) ----
  for (int i = tid; i < DOUT * DIN; i += 256) sLo[i] = Wraw[i];
  __syncthreads();

  // ---- GEMM2: Y(64x64) = WX * W^T ; B[k][n] = W[n][k]. Y kept in regs ----
  v8f y0 = {}, y1 = {};
  const int mt0 = (w >> 2), mt1 = 2 + (w >> 2), nt = (w & 3);
  const int n = nt * 16 + m;
  for (int k = 0; k < DIN; k += 4) {
    const int kk = k + 2 * half;
    v2f a, bv;
    bv[0] = sLo[n * DIN + kk];
    bv[1] = sLo[n * DIN + kk + 1];
    a[0] = sWX[(mt0 * 16 + m) * DIN + kk];
    a[1] = sWX[(mt0 * 16 + m) * DIN + kk + 1];
    y0 = wmma4(a, bv, y0);
    a[0] = sWX[(mt1 * 16 + m) * DIN + kk];
    a[1] = sWX[(mt1 * 16 + m) * DIN + kk + 1];
    y1 = wmma4(a, bv, y1);
  }
  __syncthreads();  // all WX/W reads done; pool is now free to re-purpose

  // ---- phase 3 buffers (alias pool) ----
  float* sA0 = pool;          // Ahat = (Y_sym - eps I)/nrm
  float* sS = pool + 4096;    // sign iterate
  float* sT = pool + 8192;    // S*S
  float* sU = pool + 12288;   // S*T (also hosts reduction scalars early)
  float* red = pool + 12288;  // transient, dead until first NS iteration

  for (int r = 0; r < 8; ++r) {
    sT[(mt0 * 16 + r + 8 * half) * 64 + n] = y0[r];
    sT[(mt1 * 16 + r + 8 * half) * 64 + n] = y1[r];
  }
  __syncthreads();

  // ---- A = 0.5(Y + Y^T) - eps I ; accumulate ||A||_F^2 ----
  float lsum = 0.f;
  for (int i = 0; i < 16; ++i) {
    const int idx = tid + i * 256;
    const int r = idx >> 6, c = idx & 63;
    const float v = 0.5f * (sT[r * 64 + c] + sT[c * 64 + r]) -
                    (r == c ? EPSV : 0.f);
    sA0[idx] = v;
    lsum += v * v;
  }
  for (int off = 16; off; off >>= 1) lsum += __shfl_down(lsum, off, 32);
  if (lane == 0) red[w] = lsum;
  __syncthreads();
  if (tid == 0) {
    float t = 0.f;
    for (int i = 0; i < 8; ++i) t += red[i];
    const float nr = sqrtf(t);
    red[8] = nr;
    red[9] = (nr > 1e-30f) ? 1.f / nr : 0.f;
  }
  __syncthreads();
  const float nrm = red[8];
  const float rnrm = red[9];  // held in regs; red/sU reused below
  __syncthreads();

  // ---- normalize: Ahat, S0 = A/nrm ----
  for (int i = 0; i < 16; ++i) {
    const int idx = tid + i * 256;
    const float v = sA0[idx] * rnrm;
    sA0[idx] = v;
    sS[idx] = v;
  }
  __syncthreads();

  // ---- Newton-Schulz sign iteration: S <- 0.5 S (3I - S^2) ----
  for (int it = 0; it < NS_ITERS; ++it) {
    mm64_lds(sS, sS, sT, w, m, half);  // T = S*S
    __syncthreads();
    mm64_lds(sS, sT, sU, w, m, half);  // U = S*T
    __syncthreads();
    for (int i = 0; i < 16; ++i) {
      const int idx = tid + i * 256;
      sS[idx] = 1.5f * sS[idx] - 0.5f * sU[idx];
    }
    __syncthreads();
  }

  // ---- out = 0.5*nrm*(Ahat + Ahat*S) + eps I  (P = |Ahat| fused epilogue)
  {
    float* __restrict__ outB = out + (size_t)b * 64 * 64;
    v8f acc0 = {}, acc1 = {};
    for (int k = 0; k < 64; k += 4) {
      const int kk = k + 2 * half;
      v2f a, bv;
      bv[0] = sS[kk * 64 + n];
      bv[1] = sS[(kk + 1) * 64 + n];
      a[0] = sA0[(mt0 * 16 + m) * 64 + kk];
      a[1] = sA0[(mt0 * 16 + m) * 64 + kk + 1];
      acc0 = wmma4(a, bv, acc0);
      a[0] = sA0[(mt1 * 16 + m) * 64 + kk];
      a[1] = sA0[(mt1 * 16 + m) * 64 + kk + 1];
      acc1 = wmma4(a, bv, acc1);
    }
    for (int r = 0; r < 8; ++r) {
      const int r0 = mt0 * 16 + r + 8 * half;
      const int r1 = mt1 * 16 + r + 8 * half;
      outB[r0 * 64 + n] =
          0.5f * nrm * (sA0[r0 * 64 + n] + acc0[r]) + (r0 == n ? EPSV : 0.f);
      outB[r1 * 64 + n] =
          0.5f * nrm * (sA0[r1 * 64 + n] + acc1[r]) + (r1 == n ? EPSV : 0.f);
    }
  }
}

// ---------------------------------------------------------------------------
extern "C" void kernel_launch(void* const* d_in, const int* in_sizes, int n_in,
                              void* d_out, int out_size, void* d_ws,
                              size_t ws_size, hipStream_t stream) {
  const float* X = (const float*)d_in[0];   // (4096,128,128) f32
  const float* W = (const float*)d_in[1];   // (64,128) f32
  float* out = (float*)d_out;               // (4096,64,64) f32
  float* Wst = (float*)d_ws;                // 64*128 f32 scratch

  qr_wst_kernel<<<1, 128, 0, stream>>>(W, Wst);
  birsb_kernel<<<4096, 256, 0, stream>>>(X, W, Wst, out);
  (void)in_sizes; (void)n_in; (void)out_size; (void)ws_size;
}